// AdvancedMoE_58377195487790
// MI455X (gfx1250) — compile-verified
//
#include <hip/hip_runtime.h>
#include <math.h>

typedef __attribute__((ext_vector_type(2))) float v2f;
typedef __attribute__((ext_vector_type(4))) float v4f;
typedef __attribute__((ext_vector_type(8))) float v8f;

constexpr int D  = 768;   // input dim
constexpr int H  = 256;   // hidden dim
constexpr int E  = 8;     // experts
constexpr int WAVES = 4;            // waves per workgroup (wave32)
constexpr int TPB = WAVES * 32;     // 128 threads
constexpr int MT = 16;              // tokens per wave (WMMA M)
constexpr int MBLK = WAVES * MT;    // 64 tokens per workgroup
constexpr int KC = 64;              // K-chunk of weights staged in LDS
constexpr int PR = 544;             // pair-row stride in floats (512 + 32):
                                    // shifts lane-half 16..31 by exactly 32
                                    // LDS banks -> conflict-free b64 reads
constexpr int SWF = (KC / 2) * PR;  // staging buffer floats (69632 B)

// ---------------------------------------------------------------------------
// D(16x16 f32) = A(16x4 f32) * B(4x16 f32) + C  — native f32 matrix op.
// A frag: lanes 0-15 hold {K=k,K=k+1} of row M=lane; lanes 16-31 {K=k+2,k+3}.
// B frag: VGPR0 = rows k (lanes 0-15) / k+2 (lanes 16-31); VGPR1 = k+1 / k+3.
// C/D:    VGPR r = row M=r (lanes 0-15) / M=r+8 (lanes 16-31), N = lane&15.
// ---------------------------------------------------------------------------
__device__ __forceinline__ v8f wmma_f32_k4(v2f a, v2f b, v8f c) {
  return __builtin_amdgcn_wmma_f32_16x16x4_f32(false, a, false, b, (short)0, c,
                                               false, false);
}

__device__ __forceinline__ void zero_acc(v8f* acc) {
  #pragma unroll
  for (int n = 0; n < 16; ++n)
    #pragma unroll
    for (int r = 0; r < 8; ++r) acc[n][r] = 0.0f;
}

// Cooperative copy of a KC x 256 f32 weight tile (row-major, leading dim ldw,
// rows k0.., cols col0..col0+255) into LDS, K-PAIR-INTERLEAVED:
//   sW[(k>>1)*PR + col*2 + (k&1)] = W[k][col]
// so a B fragment {row k, row k+1} at one column is a single aligned b64 read.
// Also prefetches the next K-chunk of the same weight matrix into cache.
__device__ __forceinline__ void stage_tile(const float* __restrict__ W, int ldw,
                                           int k0, int col0, float* sW, int tid,
                                           bool prefetch_next) {
  constexpr int NIT = (KC / 2) * 64;  // pair-rows x float4-column-groups
  #pragma unroll 4
  for (int i = tid; i < NIT; i += TPB) {
    int p = i >> 6;            // pair-row index (rows 2p, 2p+1)
    int c = (i & 63) << 2;     // column group base
    const float* g0 = W + (size_t)(k0 + 2 * p) * ldw + col0 + c;
    const float* g1 = g0 + ldw;
    v4f r0 = *(const v4f*)g0;
    v4f r1 = *(const v4f*)g1;
    float* dst = sW + p * PR + c * 2;
    v2f t;
    t[0] = r0[0]; t[1] = r1[0]; *(v2f*)(dst + 0) = t;
    t[0] = r0[1]; t[1] = r1[1]; *(v2f*)(dst + 2) = t;
    t[0] = r0[2]; t[1] = r1[2]; *(v2f*)(dst + 4) = t;
    t[0] = r0[3]; t[1] = r1[3]; *(v2f*)(dst + 6) = t;
    if (prefetch_next)
      __builtin_prefetch(g0 + (size_t)KC * ldw, 0, 1);  // global_prefetch_b8
  }
}

// Shared WMMA inner loop over one staged K-chunk. Unrolled x4 so the A-frag
// global loads clause up and overlap with previous steps' WMMA bursts.
__device__ __forceinline__ void mma_chunk(const float* __restrict__ arow,
                                          int k0, const float* sW, v8f* acc,
                                          int lm, int hl) {
  #pragma unroll 4
  for (int kk = 0; kk < KC; kk += 4) {
    v2f a = *(const v2f*)(arow + k0 + kk);
    const float* wb = sW + ((kk >> 1) + hl) * PR + lm * 2;
    #pragma unroll
    for (int n = 0; n < 16; ++n) {
      v2f b = *(const v2f*)(wb + n * 32);
      acc[n] = wmma_f32_k4(a, b, acc[n]);
    }
  }
}

// C[16 x 256] = A[16 x D] (global, ld = D) @ W[D x 256] (global, ld = 256)
__device__ __forceinline__ void gemm_xW(const float* __restrict__ A,
                                        const float* __restrict__ W,
                                        float* sW, v8f* acc, int lane, int tid) {
  const int lm = lane & 15, hl = lane >> 4;
  zero_acc(acc);
  const float* arow = A + (size_t)lm * D + 2 * hl;
  #pragma unroll 1
  for (int k0 = 0; k0 < D; k0 += KC) {
    __syncthreads();                 // previous chunk fully consumed
    stage_tile(W, H, k0, 0, sW, tid, k0 + KC < D);
    __syncthreads();                 // tile visible to all waves
    mma_chunk(arow, k0, sW, acc, lm, hl);
  }
}

// C[16 x 256] = A[16 x 256] (LDS, ld = H) @ W[256 x 256-slice] (global, ld=ldw)
__device__ __forceinline__ void gemm_hW(const float* __restrict__ hIn,
                                        const float* __restrict__ W, int ldw,
                                        int col0, float* sW, v8f* acc,
                                        int lane, int tid) {
  const int lm = lane & 15, hl = lane >> 4;
  zero_acc(acc);
  const float* arow = hIn + lm * H + 2 * hl;
  #pragma unroll 1
  for (int k0 = 0; k0 < H; k0 += KC) {
    __syncthreads();
    stage_tile(W, ldw, k0, col0, sW, tid, k0 + KC < H);
    __syncthreads();
    mma_chunk(arow, k0, sW, acc, lm, hl);
  }
}

// relu(acc + bias) -> per-wave LDS h buffer (row-major 16 x 256)
__device__ __forceinline__ void store_relu(const v8f* acc,
                                           const float* __restrict__ bias,
                                           float* hOut, int lane) {
  const int lm = lane & 15, hl = lane >> 4;
  #pragma unroll
  for (int n = 0; n < 16; ++n) {
    float bv = bias[n * 16 + lm];
    #pragma unroll
    for (int r = 0; r < 8; ++r) {
      float v = acc[n][r] + bv;
      hOut[(r + 8 * hl) * H + n * 16 + lm] = v > 0.f ? v : 0.f;
    }
  }
}

__global__ __launch_bounds__(TPB) void moe_fused_kernel(
    const float* __restrict__ x,
    const float* __restrict__ W1, const float* __restrict__ b1,
    const float* __restrict__ W2, const float* __restrict__ b2,
    const float* __restrict__ W3, const float* __restrict__ b3,
    const float* __restrict__ Wg1, const float* __restrict__ bg1,
    const float* __restrict__ Wg2, const float* __restrict__ bg2,
    const float* __restrict__ Wg3, const float* __restrict__ bg3,
    const float* __restrict__ Wq1, const float* __restrict__ bq1,
    const float* __restrict__ Wq2, const float* __restrict__ bq2,
    float* __restrict__ out, int Ntok) {
  __shared__ __align__(16) float sW[SWF];          // 68 KB weight staging
  __shared__ __align__(16) float sH[WAVES][MT * H];// 64 KB per-wave activations
  __shared__ float sWt[WAVES][MT * E];             // top-k combine weights

  const int tid = threadIdx.x;
  const int wave = tid >> 5, lane = tid & 31;
  const int lm = lane & 15, hl = lane >> 4;
  const int t0 = blockIdx.x * MBLK + wave * MT;    // first token of this wave

  const float* xrow = x + (size_t)t0 * D;
  float* gout   = out;                             // [N, D]
  float* gprob  = out + (size_t)Ntok * D;          // [N, E]
  float* gscore = gprob + (size_t)Ntok * E;        // [N]

  v8f acc[16];

  // ---------------- gating network (exact f32: routing must not flip) ------
  gemm_xW(xrow, Wg1, sW, acc, lane, tid);
  store_relu(acc, bg1, sH[wave], lane);
  gemm_hW(sH[wave], Wg2, H, 0, sW, acc, lane, tid);
  store_relu(acc, bg2, sH[wave], lane);
  __syncthreads();
  if (lane < MT) {
    const int m = lane;
    float lg[E];
    #pragma unroll
    for (int e = 0; e < E; ++e) lg[e] = bg3[e];
    const float* hrow = sH[wave] + m * H;
    for (int k = 0; k < H; ++k) {
      float g = hrow[k];
      #pragma unroll
      for (int e = 0; e < E; ++e) lg[e] += g * Wg3[k * E + e];
    }
    float mx = lg[0];
    #pragma unroll
    for (int e = 1; e < E; ++e) mx = fmaxf(mx, lg[e]);
    float p[E], s = 0.f;
    #pragma unroll
    for (int e = 0; e < E; ++e) { p[e] = expf(lg[e] - mx); s += p[e]; }
    float inv = 1.f / s;
    #pragma unroll
    for (int e = 0; e < E; ++e) {
      p[e] *= inv;
      gprob[(size_t)(t0 + m) * E + e] = p[e];
    }
    // top-2 (lowest index wins ties, like jax.lax.top_k)
    int i0 = 0;
    #pragma unroll
    for (int e = 1; e < E; ++e) if (p[e] > p[i0]) i0 = e;
    int i1 = (i0 == 0) ? 1 : 0;
    #pragma unroll
    for (int e = 0; e < E; ++e) if (e != i0 && p[e] > p[i1]) i1 = e;
    float s2 = p[i0] + p[i1];
    #pragma unroll
    for (int e = 0; e < E; ++e) sWt[wave][m * E + e] = 0.f;
    sWt[wave][m * E + i0] = p[i0] / s2;
    sWt[wave][m * E + i1] = p[i1] / s2;
  }

  // ---------------- geometric score ----------------------------------------
  gemm_xW(xrow, Wq1, sW, acc, lane, tid);
  store_relu(acc, bq1, sH[wave], lane);
  __syncthreads();
  if (lane < MT) {
    float s = bq2[0];
    const float* hrow = sH[wave] + lane * H;
    for (int k = 0; k < H; ++k) s += hrow[k] * Wq2[k];
    gscore[t0 + lane] = s;
  }

  // ---------------- experts: dense compute, weighted accumulate ------------
  #pragma unroll 1
  for (int e = 0; e < E; ++e) {
    gemm_xW(xrow, W1 + (size_t)e * D * H, sW, acc, lane, tid);
    store_relu(acc, b1 + e * H, sH[wave], lane);
    gemm_hW(sH[wave], W2 + (size_t)e * H * H, H, 0, sW, acc, lane, tid);
    store_relu(acc, b2 + e * H, sH[wave], lane);
    float wv[8];
    #pragma unroll
    for (int r = 0; r < 8; ++r) wv[r] = sWt[wave][(r + 8 * hl) * E + e];
    #pragma unroll 1
    for (int c = 0; c < 3; ++c) {  // 768 output cols in 3 chunks of 256
      gemm_hW(sH[wave], W3 + (size_t)e * H * D, D, c * 256, sW, acc, lane, tid);
      if (e == 0) {
        #pragma unroll
        for (int n = 0; n < 16; ++n) {
          int col = c * 256 + n * 16 + lm;
          float b3v = b3[e * D + col];
          #pragma unroll
          for (int r = 0; r < 8; ++r)
            gout[(size_t)(t0 + r + 8 * hl) * D + col] =
                wv[r] * (acc[n][r] + b3v);
        }
      } else {
        #pragma unroll
        for (int n = 0; n < 16; ++n) {
          int col = c * 256 + n * 16 + lm;
          float b3v = b3[e * D + col];
          #pragma unroll
          for (int r = 0; r < 8; ++r)
            gout[(size_t)(t0 + r + 8 * hl) * D + col] +=
                wv[r] * (acc[n][r] + b3v);
        }
      }
    }
  }
}

extern "C" void kernel_launch(void* const* d_in, const int* in_sizes, int n_in,
                              void* d_out, int out_size, void* d_ws,
                              size_t ws_size, hipStream_t stream) {
  (void)n_in; (void)out_size; (void)d_ws; (void)ws_size;
  const float* x   = (const float*)d_in[0];
  const float* W1  = (const float*)d_in[1];
  const float* b1  = (const float*)d_in[2];
  const float* W2  = (const float*)d_in[3];
  const float* b2  = (const float*)d_in[4];
  const float* W3  = (const float*)d_in[5];
  const float* b3  = (const float*)d_in[6];
  const float* Wg1 = (const float*)d_in[7];
  const float* bg1 = (const float*)d_in[8];
  const float* Wg2 = (const float*)d_in[9];
  const float* bg2 = (const float*)d_in[10];
  const float* Wg3 = (const float*)d_in[11];
  const float* bg3 = (const float*)d_in[12];
  const float* Wq1 = (const float*)d_in[13];
  const float* bq1 = (const float*)d_in[14];
  const float* Wq2 = (const float*)d_in[15];
  const float* bq2 = (const float*)d_in[16];
  float* out = (float*)d_out;

  const int Ntok = in_sizes[0] / D;   // 16384
  const int grid = Ntok / MBLK;       // 256 workgroups of 128 threads

  hipLaunchKernelGGL(moe_fused_kernel, dim3(grid), dim3(TPB), 0, stream,
                     x, W1, b1, W2, b2, W3, b3, Wg1, bg1, Wg2, bg2, Wg3, bg3,
                     Wq1, bq1, Wq2, bq2, out, Ntok);
}